// RocketConv_8005819039750
// MI455X (gfx1250) — compile-verified
//
#include <hip/hip_runtime.h>
#include <hip/hip_bf16.h>

// RocketConv on gfx1250: 84 ternary dilated-conv kernels as a WMMA f32 GEMM.
//   Y[b,k,c,t] = sum_{j=0..8} W[k,j] * X[b,c, t+(j-4)*4]   (zero padded)
// W[k,j] = +2 if j is in the k-th lexicographic 3-combination of {0..8},
// else -1.  Encoded as a 9-bit mask per kernel.
//
// One wave computes a 16(kernels) x 16(time) f32 tile via 3 chained
// v_wmma_f32_16x16x4_f32 (taps 0-3, 4-7, 8+pad) and iterates (fully
// unrolled) over 6 k-tiles (84 kernels padded to 96), reusing the same
// B (x) fragments.  Output is streamed once with nontemporal stores.

typedef __attribute__((ext_vector_type(2))) float v2f;
typedef __attribute__((ext_vector_type(8))) float v8f;

#define T_LEN 4096
#define NK    84
#define NKT   6      // ceil(84/16)
#define OUT_KSTRIDE ((size_t)16 * T_LEN)   // stride between k planes (C*T)

struct Masks { unsigned short m[96]; };

static constexpr Masks make_masks() {
    Masks ms{};
    int i = 0;
    for (int a = 0; a < 9; ++a)
        for (int b = a + 1; b < 9; ++b)
            for (int c = b + 1; c < 9; ++c)
                ms.m[i++] = (unsigned short)((1u << a) | (1u << b) | (1u << c));
    // entries 84..95 stay 0 (pad rows; weights forced to 0 below)
    return ms;
}

__constant__ Masks g_masks = make_masks();

__global__ void __launch_bounds__(256)
rocket_conv_wmma(const float* __restrict__ x, float* __restrict__ out) {
    const int lane  = threadIdx.x & 31;   // wave32
    const int wid   = threadIdx.x >> 5;   // wave in block (0..7)
    const int gwave = blockIdx.x * 8 + wid;   // 0..65535
    const int bc    = gwave >> 8;             // 0..255 = b*16 + c
    const int ttile = gwave & 255;            // 0..255
    const int t0    = ttile << 4;             // time-tile base
    const int b     = bc >> 4;
    const int cch   = bc & 15;

    const int n  = lane & 15;   // N (time) / M (kernel row) within tile
    const int kh = lane >> 4;   // lane half: selects K sub-rows {0,1} vs {2,3}

    const float* __restrict__ xrow = x + (size_t)bc * T_LEN;

    // ---- B fragments: B[K][n] = X[t0+n + (tap-4)*4], tap = g*4 + kh*2 + e.
    // Branchless: clamp address, always load (L2 hit), select 0 if OOB/pad.
    v2f bfrag[3];
#pragma unroll
    for (int g = 0; g < 3; ++g) {
#pragma unroll
        for (int e = 0; e < 2; ++e) {
            const int  j   = g * 4 + kh * 2 + e;        // tap index 0..11
            const int  ti  = t0 + n + (j - 4) * 4;
            const bool ok  = (j < 9) & ((unsigned)ti < (unsigned)T_LEN);
            int tic = ti < 0 ? 0 : (ti >= T_LEN ? T_LEN - 1 : ti);
            const float v = xrow[tic];                  // unconditional load
            bfrag[g][e] = ok ? v : 0.0f;                // v_cndmask
        }
    }

    const size_t outBase =
        (size_t)b * NK * OUT_KSTRIDE + (size_t)cch * T_LEN + (size_t)t0 + (size_t)n;

#pragma unroll
    for (int kt = 0; kt < NKT; ++kt) {
        // A fragment: A[m][K] = W[kt*16 + m][g*4 + K], m = lane&15.
        const int kg   = kt * 16 + n;                 // kernel row (< 96)
        const int mask = g_masks.m[kg];               // one u16 load / k-tile
        const bool krowv = (kt < 5) || (kg < NK);     // folds to true for kt<5

        v8f acc = {};  // f32 accumulator, zero-init
#pragma unroll
        for (int g = 0; g < 3; ++g) {
            v2f a;
#pragma unroll
            for (int e = 0; e < 2; ++e) {
                const int  j    = g * 4 + kh * 2 + e;
                const bool tapv = (g < 2) || (kh == 0 && e == 0);  // j < 9
                const bool hit  = (mask >> j) & 1;
                float w = hit ? 2.0f : -1.0f;          // branchless
                w = (krowv && tapv) ? w : 0.0f;        // pad rows/taps -> 0
                a[e] = w;
            }
            // D = A(16x4) x B(4x16) + C ; 8 args:
            // (neg_a, A, neg_b, B, c_mod, C, reuse_a, reuse_b)
            acc = __builtin_amdgcn_wmma_f32_16x16x4_f32(
                false, a, false, bfrag[g], (short)0, acc, false, false);
        }

        // D layout: VGPR r holds row M = r + 8*kh, col N = lane&15.
        // kt<5: all rows valid (guard folds away).  kt=5: only kh==0, r<4.
#pragma unroll
        for (int r = 0; r < 8; ++r) {
            const int k = kt * 16 + r + 8 * kh;
            if (kt < 5 || (kh == 0 && r < 4)) {
                if (k < NK) {   // compile-time true for kt<5; lane-true at kt=5
                    __builtin_nontemporal_store(
                        acc[r], out + outBase + (size_t)k * OUT_KSTRIDE);
                }
            }
        }
    }
}

extern "C" void kernel_launch(void* const* d_in, const int* in_sizes, int n_in,
                              void* d_out, int out_size, void* d_ws, size_t ws_size,
                              hipStream_t stream) {
    const float* x  = (const float*)d_in[0];
    float*      out = (float*)d_out;

    // 256 (B*C) rows x 256 time-tiles = 65536 waves; 8 waves (256 thr) / block.
    dim3 grid(8192), block(256);
    hipLaunchKernelGGL(rocket_conv_wmma, grid, block, 0, stream, x, out);
}